// SamplingGCN_77352361001416
// MI455X (gfx1250) — compile-verified
//
#include <hip/hip_runtime.h>

typedef float v2f __attribute__((ext_vector_type(2)));
typedef float v8f __attribute__((ext_vector_type(8)));

#define N0_NODES 150000
#define N1_NODES 25000
#define N2_NODES 2048
#define E1_EDGES 625000
#define E2_EDGES 20480
#define DFEAT    256
#define DFLAT    64
#define NCLS     10

// ---------------- degree / norm kernels ----------------

__global__ void fill_ones(float* __restrict__ p, int n) {
  int i = blockIdx.x * blockDim.x + threadIdx.x;
  if (i < n) p[i] = 1.0f;   // self-loop contributes 1 to every node's degree
}

__global__ void deg_accum(const int* __restrict__ col, int ne, float* __restrict__ deg) {
  int e = blockIdx.x * blockDim.x + threadIdx.x;
  if (e < ne) atomicAdd(&deg[col[e]], 1.0f);
}

__global__ void to_rsqrt(float* __restrict__ p, int n) {
  int i = blockIdx.x * blockDim.x + threadIdx.x;
  if (i < n) p[i] = rsqrtf(p[i]);   // deg >= 1 always (self loops)
}

// ---------------- aggregation (A-hat @ X) ----------------

// agg[c][:] = dinv[c]^2 * src[c][:]   (self-loop term; initializes the accumulator)
__global__ void agg_init(const float* __restrict__ src, const float* __restrict__ dinv,
                         float* __restrict__ agg, int ntgt) {
  int idx = blockIdx.x * blockDim.x + threadIdx.x;
  if (idx >= ntgt * 64) return;
  int c = idx >> 6;
  int j = idx & 63;
  float s = dinv[c] * dinv[c];
  float4 v = ((const float4*)src)[(size_t)c * 64 + j];
  v.x *= s; v.y *= s; v.z *= s; v.w *= s;
  ((float4*)agg)[(size_t)c * 64 + j] = v;
}

// per edge: agg[col] += dinv[row]*dinv[col] * src[row]
// blockDim = (64, 4): 64 lanes * float4 cover 256 features, 4 edges per block.
__global__ void edge_scatter(const int* __restrict__ row, const int* __restrict__ col,
                             int ne, const float* __restrict__ src,
                             const float* __restrict__ dinv, float* __restrict__ agg) {
  int e = blockIdx.x * 4 + threadIdx.y;
  if (e >= ne) return;
  int r = row[e];
  int c = col[e];
  float nrm = dinv[r] * dinv[c];
  int j = threadIdx.x;                 // 0..63
  float4 v = ((const float4*)(src + (size_t)r * DFEAT))[j];
  float* base = agg + (size_t)c * DFEAT + j * 4;
  atomicAdd(base + 0, nrm * v.x);
  atomicAdd(base + 1, nrm * v.y);
  atomicAdd(base + 2, nrm * v.z);
  atomicAdd(base + 3, nrm * v.w);
}

// ---------------- f32 WMMA GEMM: C = [relu](A @ W + bias) ----------------
// A: [M x K] row-major, W: [K x N] row-major, bias: [N], C: [M x N].
// Block = 256 threads = 8 waves; macro-tile 128(M) x 64(N).
// Each wave owns one 16(M) x 64(N) strip: 4 f32 WMMA accumulators, A fragment
// loaded once per k-step and reused 4x. W chunk (64k x 64n) staged in LDS
// (row stride 72 floats -> half-wave K-row pairs hit disjoint bank sets).
// Requires: N % 64 == 0, K % 64 == 0. M handled with arithmetic masking so
// EXEC stays all-ones at every WMMA.
#define KC   64
#define LDSW 72

__global__ void gemm_wmma_f32(const float* __restrict__ A, const float* __restrict__ W,
                              const float* __restrict__ bias, float* __restrict__ C,
                              int M, int N, int K, int relu) {
  __shared__ float lw[KC * LDSW];

  const int tid  = threadIdx.x;
  const int lane = tid & 31;
  const int wave = tid >> 5;                 // 0..7 -> M sub-tile within macro-tile
  const int n0   = blockIdx.x * 64;          // N strip
  const int m0   = blockIdx.y * 128 + wave * 16;

  const int lm = lane & 15;                  // A row / B col within 16-tile
  const int kb = (lane >> 4) << 1;           // K pair: lanes 0-15 -> {0,1}, 16-31 -> {2,3}

  int arow = m0 + lm;
  float amask = (arow < M) ? 1.0f : 0.0f;    // keep EXEC full for WMMA
  const float* aptr = A + (size_t)((arow < M) ? arow : 0) * K + kb;

  // cooperative staging indices: thread t loads rows (t/16)+16i, 4 cols at (t%16)*4
  const int sr = tid >> 4;
  const int sc = (tid & 15) << 2;

  v8f acc[4] = {};

  for (int k0 = 0; k0 < K; k0 += KC) {
    __syncthreads();
#pragma unroll
    for (int i = 0; i < 4; ++i) {
      int r = sr + (i << 4);
      float4 v = *(const float4*)(W + (size_t)(k0 + r) * N + n0 + sc);
      *(float4*)(&lw[r * LDSW + sc]) = v;    // 288B row stride keeps 16B alignment
    }
    __syncthreads();

#pragma unroll 4
    for (int kk = 0; kk < KC; kk += 4) {
      float2 av = *(const float2*)(aptr + k0 + kk);
      v2f a; a.x = av.x * amask; a.y = av.y * amask;
      const float* bl = &lw[(kk + kb) * LDSW + lm];
#pragma unroll
      for (int s = 0; s < 4; ++s) {
        v2f b;
        b.x = bl[s * 16];                    // W[k0+kk+kb  ][n0 + s*16 + lm]
        b.y = bl[LDSW + s * 16];             // W[k0+kk+kb+1][n0 + s*16 + lm]
        acc[s] = __builtin_amdgcn_wmma_f32_16x16x4_f32(false, a, false, b,
                                                       (short)0, acc[s], false, false);
      }
    }
  }

  int mbase = m0 + ((lane >> 4) << 3);       // C rows 0-7 in lanes 0-15, 8-15 in 16-31
#pragma unroll
  for (int s = 0; s < 4; ++s) {
    int colN = n0 + s * 16 + lm;
    float bv = bias[colN];
#pragma unroll
    for (int v = 0; v < 8; ++v) {
      int rowM = mbase + v;
      if (rowM < M) {
        float o = acc[s][v] + bv;
        if (relu) o = fmaxf(o, 0.0f);
        C[(size_t)rowM * N + colN] = o;
      }
    }
  }
}

// ---------------- fused output layer: [h2 | fz | last] @ Wo + bo ----------------
__global__ void out_layer(const float* __restrict__ h2, const float* __restrict__ fz,
                          const float* __restrict__ last, const float* __restrict__ Wo,
                          const float* __restrict__ bo, float* __restrict__ out) {
  int idx = blockIdx.x * blockDim.x + threadIdx.x;
  if (idx >= N2_NODES * NCLS) return;
  int r = idx / NCLS;
  int c = idx - r * NCLS;
  float acc = bo[c];
  const float* h = h2 + (size_t)r * DFEAT;
#pragma unroll 4
  for (int k = 0; k < DFEAT; ++k) acc += h[k] * Wo[k * NCLS + c];
  const float* f = fz + (size_t)r * DFLAT;
#pragma unroll 4
  for (int k = 0; k < DFLAT; ++k) acc += f[k] * Wo[(DFEAT + k) * NCLS + c];
  const float* l = last + (size_t)r * DFLAT;
#pragma unroll 4
  for (int k = 0; k < DFLAT; ++k) acc += l[k] * Wo[(DFEAT + DFLAT + k) * NCLS + c];
  out[idx] = acc;
}

// ---------------- host launch ----------------

extern "C" void kernel_launch(void* const* d_in, const int* in_sizes, int n_in,
                              void* d_out, int out_size, void* d_ws, size_t ws_size,
                              hipStream_t stream) {
  const float* x    = (const float*)d_in[0];
  const float* flat = (const float*)d_in[1];
  const float* last = (const float*)d_in[2];
  const float* W1   = (const float*)d_in[3];
  const float* b1   = (const float*)d_in[4];
  const float* W2   = (const float*)d_in[5];
  const float* b2   = (const float*)d_in[6];
  const float* Wf   = (const float*)d_in[7];
  const float* bf   = (const float*)d_in[8];
  const float* Wo   = (const float*)d_in[9];
  const float* bo   = (const float*)d_in[10];
  const int*   e1   = (const int*)d_in[11];   // [2, E1]: rows then cols
  const int*   e2   = (const int*)d_in[12];   // [2, E2]
  // d_in[13]=n1, d_in[14]=n2: static shapes (25000, 2048), baked in.

  float* ws    = (float*)d_ws;
  float* dinv1 = ws;                         ws += N0_NODES;
  float* dinv2 = ws;                         ws += N1_NODES;
  float* agg1  = ws;                         ws += (size_t)N1_NODES * DFEAT;
  float* h1    = ws;                         ws += (size_t)N1_NODES * DFEAT;
  float* agg2  = ws;                         ws += (size_t)N2_NODES * DFEAT;
  float* h2    = ws;                         ws += (size_t)N2_NODES * DFEAT;
  float* fz    = ws;                         // 2048 x 64

  // 1) degrees (self-loop => init 1.0), then dinv = rsqrt(deg)
  fill_ones<<<(N0_NODES + 255) / 256, 256, 0, stream>>>(dinv1, N0_NODES);
  fill_ones<<<(N1_NODES + 255) / 256, 256, 0, stream>>>(dinv2, N1_NODES);
  deg_accum<<<(E1_EDGES + 255) / 256, 256, 0, stream>>>(e1 + E1_EDGES, E1_EDGES, dinv1);
  deg_accum<<<(E2_EDGES + 255) / 256, 256, 0, stream>>>(e2 + E2_EDGES, E2_EDGES, dinv2);
  to_rsqrt<<<(N0_NODES + 255) / 256, 256, 0, stream>>>(dinv1, N0_NODES);
  to_rsqrt<<<(N1_NODES + 255) / 256, 256, 0, stream>>>(dinv2, N1_NODES);

  // 2) hop 1: agg1 = A-hat1 @ x (targets < n1), then h1 = relu(agg1 @ W1 + b1)
  agg_init<<<(N1_NODES * 64 + 255) / 256, 256, 0, stream>>>(x, dinv1, agg1, N1_NODES);
  edge_scatter<<<(E1_EDGES + 3) / 4, dim3(64, 4), 0, stream>>>(e1, e1 + E1_EDGES, E1_EDGES,
                                                               x, dinv1, agg1);
  gemm_wmma_f32<<<dim3(DFEAT / 64, (N1_NODES + 127) / 128), 256, 0, stream>>>(
      agg1, W1, b1, h1, N1_NODES, DFEAT, DFEAT, 1);

  // 3) hop 2: agg2 = A-hat2 @ h1 (targets < n2), then h2 = agg2 @ W2 + b2
  agg_init<<<(N2_NODES * 64 + 255) / 256, 256, 0, stream>>>(h1, dinv2, agg2, N2_NODES);
  edge_scatter<<<(E2_EDGES + 3) / 4, dim3(64, 4), 0, stream>>>(e2, e2 + E2_EDGES, E2_EDGES,
                                                               h1, dinv2, agg2);
  gemm_wmma_f32<<<dim3(DFEAT / 64, (N2_NODES + 127) / 128), 256, 0, stream>>>(
      agg2, W2, b2, h2, N2_NODES, DFEAT, DFEAT, 0);

  // 4) flat path: fz = flat @ Wf + bf  (2048x64 @ 64x64)
  gemm_wmma_f32<<<dim3(DFLAT / 64, (N2_NODES + 127) / 128), 256, 0, stream>>>(
      flat, Wf, bf, fz, N2_NODES, DFLAT, DFLAT, 0);

  // 5) fused concat + output GEMM (N=10, VALU kernel)
  out_layer<<<(N2_NODES * NCLS + 255) / 256, 256, 0, stream>>>(h2, fz, last, Wo, bo,
                                                               (float*)d_out);
}